// SuperpixelSAN_81123342287661
// MI455X (gfx1250) — compile-verified
//
#include <hip/hip_runtime.h>
#include <hip/hip_bf16.h>

#define FD 18      // F_SIZE
#define LDH 20     // padded row stride for h (80B -> 16B-aligned rows)
#define OUTD 10    // OUT_SIZE
#define NG 256     // NUM_GRAPHS

typedef __attribute__((ext_vector_type(2))) float v2f;
typedef __attribute__((ext_vector_type(8))) float v8f;

// ---------------------------------------------------------------------------
// Pack W[K,N] (+zero pad) into per-lane WMMA B layout:
//   Bp[((kt*ntile + nt)*32 + lane)*2 + {0,1}] = W[kt*4 + (lane>>4)*2 + {0,1}][nt*16 + (lane&15)]
// so the GEMM kernel does one unconditional 8B load per lane per K-step.
// ---------------------------------------------------------------------------
__global__ void pack_w_kernel(const float* __restrict__ W, float* __restrict__ Bp,
                              int K, int N, int K4, int ntile) {
  int p = blockIdx.x * blockDim.x + threadIdx.x;
  int tot = K4 * ntile * 32;
  if (p >= tot) return;
  int kt   = p / (ntile * 32);
  int rem  = p - kt * ntile * 32;
  int nt   = rem >> 5;
  int lane = rem & 31;
  int k    = kt * 4 + (lane >> 4) * 2;
  int col  = nt * 16 + (lane & 15);
  Bp[p * 2 + 0] = (k     < K && col < N) ? W[k * N + col]       : 0.0f;
  Bp[p * 2 + 1] = (k + 1 < K && col < N) ? W[(k + 1) * N + col] : 0.0f;
}

__global__ void pack_bias_kernel(const float* __restrict__ bias, float* __restrict__ bp,
                                 int N, int Npad) {
  int i = threadIdx.x;
  if (i < Npad) bp[i] = (i < N) ? bias[i] : 0.0f;
}

// zero-pad X[n,feat] -> Xp[n,Kpad] so layer-1 A reads are unguarded & aligned
__global__ void pad_x_kernel(const float* __restrict__ X, float* __restrict__ Xp,
                             int n, int feat, int Kpad) {
  int i = blockIdx.x * blockDim.x + threadIdx.x;
  if (i >= n * Kpad) return;
  int r = i / Kpad, k = i - r * Kpad;
  Xp[i] = (k < feat) ? X[r * feat + k] : 0.0f;
}

// ---------------------------------------------------------------------------
// f32 WMMA linear: O[M,N] = A[M,K]@W + bias.  One wave32 per 16x16 tile,
// V_WMMA_F32_16X16X4_F32, K stepped by 4.  M is always a multiple of 16 here;
// K/N padding handled by zero-padded Bp/biasp (A over-reads stay in d_ws).
// A layout per ISA "32-bit A-Matrix 16x4": lanes0-15 K=k0,k0+1; lanes16-31
// K=k0+2,k0+3; M=lane&15.  D: VGPR r -> row r (+8 upper half), col=lane&15.
// ---------------------------------------------------------------------------
__global__ void wmma_linear_kernel(const float* __restrict__ A, int lda,
                                   const float* __restrict__ Bp,
                                   const float* __restrict__ biasp,
                                   float* __restrict__ O, int ldo,
                                   int M, int K4, int ntile, int N) {
  int wave = (blockIdx.x * blockDim.x + threadIdx.x) >> 5;
  int lane = threadIdx.x & 31;
  int mt = wave / ntile, nt = wave - mt * ntile;
  int m0 = mt << 4, n0 = nt << 4;
  if (m0 >= M) return;                 // wave-uniform; EXEC all-1s inside
  int half = lane >> 4, l16 = lane & 15;

  const float* Ap = A + (size_t)(m0 + l16) * lda + half * 2;  // 8B aligned (lda even)
  const float* Bq = Bp + nt * 64 + lane * 2;
  int bstride = ntile * 64;

  v8f c = {};
  for (int kt = 0; kt < K4; ++kt) {
    v2f a = *(const v2f*)(Ap + kt * 4);
    v2f b = *(const v2f*)(Bq + kt * bstride);
    c = __builtin_amdgcn_wmma_f32_16x16x4_f32(false, a, false, b,
                                              (short)0, c, false, false);
  }

  int col = n0 + l16;
  if (col < N) {
    float bb = biasp[col];
    float* Ob = O + (size_t)m0 * ldo + (size_t)(half * 8) * ldo + col;
#pragma unroll
    for (int r = 0; r < 8; ++r) Ob[(size_t)r * ldo] = c[r] + bb;
  }
}

// --------------------------- elementwise helpers ---------------------------
__global__ void fill_kernel(float* p, float v, int n) {
  int i = blockIdx.x * blockDim.x + threadIdx.x;
  if (i < n) p[i] = v;
}

__global__ void relu_store_kernel(const float* __restrict__ acc,
                                  float* __restrict__ xcat, int col_off, int n) {
  int i = blockIdx.x * blockDim.x + threadIdx.x;
  if (i >= n) return;
  const float* a = acc + (size_t)i * FD;
  float* o = xcat + (size_t)i * (3 * FD) + col_off;
#pragma unroll
  for (int f = 0; f < FD; ++f) o[f] = fmaxf(a[f], 0.0f);
}

// vectorized 18-float row gather from h (row stride LDH=20, 16B-aligned rows)
__device__ __forceinline__ void load_row18(const float* __restrict__ h, int col,
                                           float v[FD]) {
  const float* p = h + (size_t)col * LDH;
  float4 q0 = *(const float4*)(p);
  float4 q1 = *(const float4*)(p + 4);
  float4 q2 = *(const float4*)(p + 8);
  float4 q3 = *(const float4*)(p + 12);
  float2 q4 = *(const float2*)(p + 16);
  v[0]=q0.x; v[1]=q0.y; v[2]=q0.z; v[3]=q0.w;
  v[4]=q1.x; v[5]=q1.y; v[6]=q1.z; v[7]=q1.w;
  v[8]=q2.x; v[9]=q2.y; v[10]=q2.z; v[11]=q2.w;
  v[12]=q3.x; v[13]=q3.y; v[14]=q3.z; v[15]=q3.w;
  v[16]=q4.x; v[17]=q4.y;
}

// ------------------------------ sparse ops ---------------------------------
__global__ void spmm_kernel(const int* __restrict__ rows, const int* __restrict__ cols,
                            const float* __restrict__ val, const float* __restrict__ h,
                            float* __restrict__ acc, int nnz) {
  int i = blockIdx.x * blockDim.x + threadIdx.x;
  if (i >= nnz) return;
  float hv[FD];
  load_row18(h, cols[i], hv);
  float v = val[i];
  float* ar = acc + (size_t)rows[i] * FD;
#pragma unroll
  for (int f = 0; f < FD; ++f) atomicAdd(&ar[f], v * hv[f]);
}

__global__ void ha_kernel(const float* __restrict__ h, const float* __restrict__ a1,
                          const float* __restrict__ a2,
                          float* __restrict__ hA1, float* __restrict__ hA2, int n) {
  int i = blockIdx.x * blockDim.x + threadIdx.x;
  if (i >= n) return;
  float hv[FD];
  load_row18(h, i, hv);
  float s1 = 0.0f, s2 = 0.0f;
#pragma unroll
  for (int f = 0; f < FD; ++f) { s1 += hv[f] * a1[f]; s2 += hv[f] * a2[f]; }
  hA1[i] = s1; hA2[i] = s2;
}

__device__ __forceinline__ void atomicMaxF32(float* addr, float v) {
  if (v >= 0.0f) atomicMax((int*)addr, __float_as_int(v));
  else           atomicMin((unsigned int*)addr, __float_as_uint(v));
}

__global__ void gat_edge1_kernel(const int* __restrict__ rows, const int* __restrict__ cols,
                                 const float* __restrict__ hA1, const float* __restrict__ hA2,
                                 float* __restrict__ ebuf, float* __restrict__ segmax, int nnz) {
  int i = blockIdx.x * blockDim.x + threadIdx.x;
  if (i >= nnz) return;
  float e = hA1[rows[i]] + hA2[cols[i]];
  e = (e > 0.0f) ? e : 0.2f * e;
  ebuf[i] = e;
  atomicMaxF32(&segmax[rows[i]], e);
}

__global__ void gat_edge2_kernel(const int* __restrict__ rows,
                                 float* __restrict__ ebuf,
                                 const float* __restrict__ segmax,
                                 float* __restrict__ denom, int nnz) {
  int i = blockIdx.x * blockDim.x + threadIdx.x;
  if (i >= nnz) return;
  float m = segmax[rows[i]];
  if (isinf(m)) m = 0.0f;             // empty rows: segment_max = -inf -> 0
  float ex = __expf(ebuf[i] - m);
  ebuf[i] = ex;
  atomicAdd(&denom[rows[i]], ex);
}

__global__ void gat_edge3_kernel(const int* __restrict__ rows, const int* __restrict__ cols,
                                 const float* __restrict__ ebuf,
                                 const float* __restrict__ denom,
                                 const float* __restrict__ h,
                                 float* __restrict__ acc, int nnz) {
  int i = blockIdx.x * blockDim.x + threadIdx.x;
  if (i >= nnz) return;
  int r = rows[i];
  float alpha = ebuf[i] / (denom[r] + 1e-16f);
  float hv[FD];
  load_row18(h, cols[i], hv);
  float* ar = acc + (size_t)r * FD;
#pragma unroll
  for (int f = 0; f < FD; ++f) atomicAdd(&ar[f], alpha * hv[f]);
}

// ------------------------------- pooling -----------------------------------
__global__ void pool_kernel(const float* __restrict__ x4, const int* __restrict__ batch,
                            float* __restrict__ pooled /* base+10t, row stride 30 */,
                            float* __restrict__ cnt, int n) {
  int i = blockIdx.x * blockDim.x + threadIdx.x;
  if (i >= n) return;
  int g = batch[i];
  atomicAdd(&cnt[g], 1.0f);
  const float* xi = x4 + (size_t)i * OUTD;
#pragma unroll
  for (int f = 0; f < OUTD; ++f) atomicAdd(&pooled[g * 30 + f], xi[f]);
}

__global__ void pool_div_kernel(float* __restrict__ pooled, const float* __restrict__ cnt) {
  int i = blockIdx.x * blockDim.x + threadIdx.x;
  if (i >= NG * OUTD) return;
  int g = i / OUTD, f = i - g * OUTD;
  pooled[g * 30 + f] /= fmaxf(cnt[g], 1.0f);
}

__global__ void softmax_kernel(const float* __restrict__ logits, float* __restrict__ out) {
  int g = blockIdx.x * blockDim.x + threadIdx.x;
  if (g >= NG) return;
  const float* l = logits + g * OUTD;
  float m = l[0];
#pragma unroll
  for (int f = 1; f < OUTD; ++f) m = fmaxf(m, l[f]);
  float e[OUTD], s = 0.0f;
#pragma unroll
  for (int f = 0; f < OUTD; ++f) { e[f] = __expf(l[f] - m); s += e[f]; }
#pragma unroll
  for (int f = 0; f < OUTD; ++f) out[g * OUTD + f] = e[f] / s;
}

// ------------------------------- host side ---------------------------------
static inline void launch_linear(const float* A, int lda, const float* W, const float* bias,
                                 float* O, int ldo, int M, int K, int N,
                                 float* wpack, float* bpack, hipStream_t s) {
  int K4 = (K + 3) >> 2, ntile = (N + 15) >> 4;
  int pairs = K4 * ntile * 32;
  pack_w_kernel<<<(pairs + 255) / 256, 256, 0, s>>>(W, wpack, K, N, K4, ntile);
  pack_bias_kernel<<<1, 32, 0, s>>>(bias, bpack, N, ntile * 16);
  long long waves = (long long)(M >> 4) * ntile;     // M is multiple of 16
  int threads = 256;                                 // 8 waves per block
  long long blocks = (waves * 32 + threads - 1) / threads;
  wmma_linear_kernel<<<(unsigned)blocks, threads, 0, s>>>(A, lda, wpack, bpack,
                                                          O, ldo, M, K4, ntile, N);
}

struct TrackCfg {
  const float* X; int n; int feat;
  const int* p_idx; const float* p_val; int nnz_p;
  const int* lu; int nnz_lu;
  const int* ld; int nnz_ld;
  const int* batch;
  int pbase;
};

extern "C" void kernel_launch(void* const* d_in, const int* in_sizes, int n_in,
                              void* d_out, int out_size, void* d_ws, size_t ws_size,
                              hipStream_t stream) {
  const int N0 = 262144, N1 = 524288, N2 = 131072;
  const int NNZ0 = 2097152, NNZ1 = 4194304, NNZ2 = 1048576;
  const int NNZ1U = 2097152, NNZ1D = 2097152;

  // Param leaves (recursive dict-insertion order):
  //   14..: t0{ l1{pW,pb,ldW,ldb,a1,a2} l2 l3 W4 b4 } (20), 34..: t1, 54..: t2,
  //   74: cW, 75: cb
  auto Pf = [&](int i) { return (const float*)d_in[i]; };

  TrackCfg tr[3];
  tr[0] = { (const float*)d_in[0], N0, 3,
            (const int*)d_in[3], (const float*)d_in[4], NNZ0,
            nullptr, 0, (const int*)d_in[3], NNZ0,
            (const int*)d_in[11], 14 };
  tr[1] = { (const float*)d_in[1], N1, 6,
            (const int*)d_in[5], (const float*)d_in[6], NNZ1,
            (const int*)d_in[9], NNZ1U, (const int*)d_in[10], NNZ1D,
            (const int*)d_in[12], 34 };
  tr[2] = { (const float*)d_in[2], N2, 9,
            (const int*)d_in[7], (const float*)d_in[8], NNZ2,
            (const int*)d_in[7], NNZ2, nullptr, 0,
            (const int*)d_in[13], 54 };

  // Workspace layout (floats), sized for the largest track (n = 524288)
  const size_t NMAX = 524288;
  float* ws     = (float*)d_ws;
  float* xcat   = ws;                         // 54*NMAX  [x1|x2|x3] per node
  float* h      = xcat   + 54 * NMAX;         // 20*NMAX  (16B-aligned rows)
  float* acc    = h      + (size_t)LDH * NMAX;// 18*NMAX
  float* hA1    = acc    + 18 * NMAX;
  float* hA2    = hA1    + NMAX;
  float* segmax = hA2    + NMAX;
  float* denom  = segmax + NMAX;
  float* ebuf   = denom  + NMAX;              // 4M edge scratch
  float* x4     = ebuf   + 4194304;           // 10*NMAX
  float* xpad   = x4     + 10 * NMAX;         // 12*NMAX padded layer-1 input
  float* pooled = xpad   + 12 * NMAX;         // 256*30
  float* cnt    = pooled + NG * 30;           // 256
  float* logits = cnt    + NG;                // 256*10
  float* wpack  = logits + NG * OUTD;         // <= 14*2*64 = 1792
  float* bpack  = wpack  + 2048;              // <= 32

  const int TB = 256;
  hipMemsetAsync(pooled, 0, (size_t)NG * 30 * sizeof(float), stream);

  for (int t = 0; t < 3; ++t) {
    const TrackCfg& c = tr[t];
    int nB = (c.n + TB - 1) / TB;

    // layer-1 input, zero-padded to 4-multiple K for unguarded WMMA A loads
    int Kpad0 = ((c.feat + 3) >> 2) << 2;
    pad_x_kernel<<<(c.n * Kpad0 + TB - 1) / TB, TB, 0, stream>>>(c.X, xpad, c.n, c.feat, Kpad0);

    for (int l = 0; l < 3; ++l) {
      int sb = c.pbase + l * 6;
      const float *pW = Pf(sb), *pb = Pf(sb + 1), *ldW = Pf(sb + 2),
                  *ldb = Pf(sb + 3), *a1 = Pf(sb + 4), *a2 = Pf(sb + 5);
      const float* in = (l == 0) ? xpad : (xcat + FD * (l - 1));
      int lda = (l == 0) ? Kpad0 : 3 * FD;
      int K   = (l == 0) ? c.feat : FD;

      hipMemsetAsync(acc, 0, (size_t)c.n * FD * sizeof(float), stream);

      // SpMM branch: h = x@pW+pb ; acc += L @ h
      launch_linear(in, lda, pW, pb, h, LDH, c.n, K, FD, wpack, bpack, stream);
      {
        int eB = (c.nnz_p + TB - 1) / TB;
        spmm_kernel<<<eB, TB, 0, stream>>>(c.p_idx, c.p_idx + c.nnz_p, c.p_val, h, acc, c.nnz_p);
      }

      // GAT branches: linear + attention dots computed ONCE (shared ldW/a1/a2)
      if (c.lu || c.ld) {
        launch_linear(in, lda, ldW, ldb, h, LDH, c.n, K, FD, wpack, bpack, stream);
        ha_kernel<<<nB, TB, 0, stream>>>(h, a1, a2, hA1, hA2, c.n);

        const int* sets[2] = { c.lu, c.ld };
        const int  nnzs[2] = { c.nnz_lu, c.nnz_ld };
        for (int si = 0; si < 2; ++si) {
          if (!sets[si]) continue;
          int nnz = nnzs[si];
          const int* rows = sets[si];
          const int* cols = sets[si] + nnz;
          int eB = (nnz + TB - 1) / TB;
          fill_kernel<<<nB, TB, 0, stream>>>(segmax, -INFINITY, c.n);
          hipMemsetAsync(denom, 0, (size_t)c.n * sizeof(float), stream);
          gat_edge1_kernel<<<eB, TB, 0, stream>>>(rows, cols, hA1, hA2, ebuf, segmax, nnz);
          gat_edge2_kernel<<<eB, TB, 0, stream>>>(rows, ebuf, segmax, denom, nnz);
          gat_edge3_kernel<<<eB, TB, 0, stream>>>(rows, cols, ebuf, denom, h, acc, nnz);
        }
      }
      relu_store_kernel<<<nB, TB, 0, stream>>>(acc, xcat, FD * l, c.n);
    }

    // x4 = concat(x1,x2,x3) @ W4 + b4
    launch_linear(xcat, 3 * FD, Pf(c.pbase + 18), Pf(c.pbase + 19),
                  x4, OUTD, c.n, 3 * FD, OUTD, wpack, bpack, stream);

    // mean pool into pooled[:, 10*t : 10*t+10]
    hipMemsetAsync(cnt, 0, NG * sizeof(float), stream);
    pool_kernel<<<nB, TB, 0, stream>>>(x4, c.batch, pooled + OUTD * t, cnt, c.n);
    pool_div_kernel<<<(NG * OUTD + TB - 1) / TB, TB, 0, stream>>>(pooled + OUTD * t, cnt);
  }

  // classifier + softmax
  launch_linear(pooled, 30, Pf(74), Pf(75), logits, OUTD, NG, 30, OUTD,
                wpack, bpack, stream);
  softmax_kernel<<<(NG + TB - 1) / TB, TB, 0, stream>>>(logits, (float*)d_out);
}